// AdjacencyMatrixLayer_73186242724237
// MI455X (gfx1250) — compile-verified
//
#include <hip/hip_runtime.h>

typedef __attribute__((ext_vector_type(2))) float v2f;
typedef __attribute__((ext_vector_type(8))) float v8f;

#define EPSN 1e-20f
#define LOG2E 1.44269504088896340736f

// One wave (32 lanes) owns a 16-row tile of the [N,N] output for one batch.
// WMMA f32 16x16x4 computes the Gram tile (K = D=3 padded to 4).
__global__ __launch_bounds__(256) void AdjacencyMatrixLayer_kernel(
    const float* __restrict__ coords,   // [B, N, 3]
    const float* __restrict__ masks,    // [B, N, N] (only diagonal is read)
    const float* __restrict__ sigma,    // [1]
    float* __restrict__ out,            // [B, N, N]
    int B, int N)
{
    const int lane = threadIdx.x & 31;
    const int wave = threadIdx.x >> 5;
    const int tilesPerBatch = N >> 4;
    const int tile = blockIdx.x * (blockDim.x >> 5) + wave;
    const int b = tile / tilesPerBatch;
    if (b >= B) return;                       // uniform per wave: EXEC stays all-1s
    const int i0 = (tile - b * tilesPerBatch) << 4;

    const float s = sigma[0];
    const float inv_sig2 = 1.0f / (s * s);

    const float* cb = coords + (size_t)b * N * 3;
    const float* mb = masks  + (size_t)b * N * N;
    float*       ob = out    + (size_t)b * N * N;

    const int  lrow = lane & 15;              // 0..15 within half-wave
    const bool hi   = lane >= 16;

    // ---- i-tile: A fragment (16x4, K3 padded with 0) + per-row norm/valid ----
    const int irow = i0 + lrow;
    const float xi = cb[irow * 3 + 0];
    const float yi = cb[irow * 3 + 1];
    const float zi = cb[irow * 3 + 2];
    const float nI = xi * xi + yi * yi + zi * zi;
    const float vI = mb[(size_t)irow * N + irow];   // mask diagonal == valid[i]

    v2f afrag;
    afrag.x = hi ? zi : xi;                   // lanes 0-15: K=0 ; lanes 16-31: K=2
    afrag.y = hi ? 0.0f : yi;                 // lanes 0-15: K=1 ; lanes 16-31: K=3 (pad)

    // C/D layout: VGPR r -> row r (lanes 0-15) / row r+8 (lanes 16-31).
    float ni[8], vi[8];
    #pragma unroll
    for (int r = 0; r < 8; ++r) {
        const int src = r + (hi ? 8 : 0);
        ni[r] = __shfl(nI, src, 32);
        vi[r] = __shfl(vI, src, 32);
    }

    // ---- pass 1: per-row S = sum(e), T = sum(e * valid_j) ----
    float S[8], T[8];
    #pragma unroll
    for (int r = 0; r < 8; ++r) { S[r] = 0.0f; T[r] = 0.0f; }

    for (int j0 = 0; j0 < N; j0 += 16) {
        const int jrow = j0 + lrow;
        const float xj = cb[jrow * 3 + 0];
        const float yj = cb[jrow * 3 + 1];
        const float zj = cb[jrow * 3 + 2];
        const float nj = xj * xj + yj * yj + zj * zj;
        const float vj = mb[(size_t)jrow * N + jrow];

        v2f bfrag;
        bfrag.x = hi ? zj : xj;
        bfrag.y = hi ? 0.0f : yj;

        v8f acc = {};
        acc = __builtin_amdgcn_wmma_f32_16x16x4_f32(
            false, afrag, false, bfrag, (short)0, acc, false, false);

        #pragma unroll
        for (int r = 0; r < 8; ++r) {
            // A = exp(-(ni + nj - 2*gram)/sig^2); arg <= ~0 so exp2 is safe
            const float arg = (2.0f * acc[r] - ni[r] - nj) * inv_sig2;
            const float e = __builtin_amdgcn_exp2f(arg * LOG2E); // gaussian, in (0,1]
            const float ee = __builtin_amdgcn_exp2f(e * LOG2E);  // softmax numerator exp(A)
            S[r] += ee;
            T[r] += ee * vj;
        }
    }

    // Butterfly reduction across the 16 lanes of each half (rows r / r+8).
    #pragma unroll
    for (int r = 0; r < 8; ++r) {
        #pragma unroll
        for (int off = 1; off < 16; off <<= 1) {
            S[r] += __shfl_xor(S[r], off, 32);
            T[r] += __shfl_xor(T[r], off, 32);
        }
    }

    // out = e * v_j * v_i / (T + EPS * S)   (softmax denom cancels)
    float scale[8];
    #pragma unroll
    for (int r = 0; r < 8; ++r)
        scale[r] = vi[r] / (T[r] + EPSN * S[r]);

    // Precompute output row bases for this lane's 8 rows.
    float* orow[8];
    #pragma unroll
    for (int r = 0; r < 8; ++r)
        orow[r] = ob + (size_t)(i0 + r + (hi ? 8 : 0)) * N;

    // ---- pass 2: recompute tiles and stream results (non-temporal stores) ----
    for (int j0 = 0; j0 < N; j0 += 16) {
        const int jrow = j0 + lrow;
        const float xj = cb[jrow * 3 + 0];
        const float yj = cb[jrow * 3 + 1];
        const float zj = cb[jrow * 3 + 2];
        const float nj = xj * xj + yj * yj + zj * zj;
        const float vj = mb[(size_t)jrow * N + jrow];

        v2f bfrag;
        bfrag.x = hi ? zj : xj;
        bfrag.y = hi ? 0.0f : yj;

        v8f acc = {};
        acc = __builtin_amdgcn_wmma_f32_16x16x4_f32(
            false, afrag, false, bfrag, (short)0, acc, false, false);

        #pragma unroll
        for (int r = 0; r < 8; ++r) {
            const float arg = (2.0f * acc[r] - ni[r] - nj) * inv_sig2;
            const float e  = __builtin_amdgcn_exp2f(arg * LOG2E);
            const float ee = __builtin_amdgcn_exp2f(e * LOG2E);
            // streamed once, never re-read -> NT hint keeps GL2 for coords/diag
            __builtin_nontemporal_store(ee * vj * scale[r], &orow[r][jrow]);
        }
    }
}

extern "C" void kernel_launch(void* const* d_in, const int* in_sizes, int n_in,
                              void* d_out, int out_size, void* d_ws, size_t ws_size,
                              hipStream_t stream) {
    const float* coords = (const float*)d_in[0];
    const float* masks  = (const float*)d_in[1];
    const float* sigma  = (const float*)d_in[2];
    float* out = (float*)d_out;

    // in_sizes[0] = B*N*3, in_sizes[1] = B*N*N  ->  N = 3*s1/s0, B = s0/(3N)
    const long long s0 = in_sizes[0];
    const long long s1 = in_sizes[1];
    const int N = (int)((3LL * s1) / s0);
    const int B = (int)(s0 / (3LL * (long long)N));

    const int tiles = B * (N >> 4);           // one wave per 16-row tile
    const int wavesPerBlock = 8;              // 256 threads
    dim3 block(32 * wavesPerBlock);
    dim3 grid((tiles + wavesPerBlock - 1) / wavesPerBlock);
    AdjacencyMatrixLayer_kernel<<<grid, block, 0, stream>>>(coords, masks, sigma, out, B, N);
}